// Resizer_39934605918907
// MI455X (gfx1250) — compile-verified
//
#include <hip/hip_runtime.h>
#include <hip/hip_bf16.h>

// Fused separable bicubic resize (32,3,512,512) -> (32,3,256,256), fp32.
// Per block: one 16x16 output tile of one (n,c) plane.
//   OUT(16x16) = Hw(16x48) * Patch(48x48) * WwT(48x16)
// via V_WMMA_F32_16X16X4_F32 (exact fp32 accumulate, RNE).
// Mirror-boundary taps become duplicate fov indices -> folded by LDS atomicAdd
// when densifying the weight windows.
// Patch is DMA'd global->LDS with GLOBAL_LOAD_ASYNC_TO_LDS_B32 (ASYNCcnt),
// overlapped with weight densification; falls back to synchronous loads if the
// async builtins are unavailable on this toolchain.

typedef __attribute__((ext_vector_type(2))) float v2f;
typedef __attribute__((ext_vector_type(8))) float v8f;

#define IN_HW   512
#define OUT_HW  256
#define PLANES  96        // 32 * 3
#define TILE    16
#define RK      48        // padded K window (input rows/cols feeding one tile)
#define HSTRIDE 49        // padded LDS row stride (49*lm mod 64 hits distinct banks)

#if defined(__has_builtin)
#  if __has_builtin(__builtin_amdgcn_global_load_async_to_lds_b32) && \
      __has_builtin(__builtin_amdgcn_s_wait_asynccnt)
#    define USE_ASYNC_LDS 1
#  endif
#endif

#ifdef USE_ASYNC_LDS
typedef __attribute__((address_space(1))) int gas_int;  // global (AS1) int
typedef __attribute__((address_space(3))) int las_int;  // LDS (AS3) int
#endif

__device__ __forceinline__ v8f wmma_f32(v2f a, v2f b, v8f c) {
  // 8-arg pattern: (neg_a, A, neg_b, B, c_mod, C, reuse_a, reuse_b)
  return __builtin_amdgcn_wmma_f32_16x16x4_f32(false, a, false, b, (short)0, c,
                                               false, false);
}

__global__ __launch_bounds__(32)
void resize_bicubic_fused(const float* __restrict__ in,
                          const float* __restrict__ w2,
                          const int*   __restrict__ fov2,
                          const float* __restrict__ w3,
                          const int*   __restrict__ fov3,
                          float*       __restrict__ out,
                          int taps)
{
  __shared__ float Hw[TILE * HSTRIDE];   // dense H weights, A of pass 1 (16 x 48)
  __shared__ float WwT[RK * TILE];       // dense W weights^T, B of pass 2 (48 x 16)
  __shared__ float P[RK * RK];           // input patch (rows x cols)
  __shared__ float T[TILE * HSTRIDE];    // temp = Hw @ P (16 x 48)

  const int lane = threadIdx.x;          // wave32: one wave per block
  const int g    = lane >> 4;            // lane group 0/1 -> K halves
  const int lm   = lane & 15;

  const int bid = blockIdx.x;
  const int tw  = bid & 15;
  const int th  = (bid >> 4) & 15;
  const int p   = bid >> 8;              // plane 0..95
  const int oh0 = th * TILE;
  const int ow0 = tw * TILE;

  // ---- window minima r0 (rows) / c0 (cols): wave shuffle reduction ----
  int mnr = 0x7fffffff, mnc = 0x7fffffff;
  for (int idx = lane; idx < taps * TILE; idx += 32) {
    const int t = idx >> 4, j = idx & 15;
    mnr = min(mnr, fov2[t * OUT_HW + oh0 + j]);
    mnc = min(mnc, fov3[t * OUT_HW + ow0 + j]);
  }
  #pragma unroll
  for (int off = 16; off > 0; off >>= 1) {
    mnr = min(mnr, __shfl_xor(mnr, off, 32));
    mnc = min(mnc, __shfl_xor(mnc, off, 32));
  }
  const int r0 = mnr, c0 = mnc;          // >= 0; tap span <= 2*15 + taps <= 48

  // ---- kick off patch DMA ASAP (overlaps with weight densification) ----
  const float* inp = in + (size_t)p * IN_HW * IN_HW;
#ifdef USE_ASYNC_LDS
  for (int idx = lane; idx < RK * RK; idx += 32) {
    const int rr = idx / RK, cc = idx - rr * RK;
    // Clamp past-edge taps to the last valid texel: their densified weights
    // are exactly 0.0f (never accumulated), and finite * 0 == 0.
    const int r = min(r0 + rr, IN_HW - 1);
    const int c = min(c0 + cc, IN_HW - 1);
    __builtin_amdgcn_global_load_async_to_lds_b32(
        (gas_int*)(inp + r * IN_HW + c),
        (las_int*)&P[idx],
        0, 0);
  }
#endif

  // ---- zero the dense weight windows ----
  for (int i = lane; i < TILE * HSTRIDE; i += 32) Hw[i] = 0.f;
  for (int i = lane; i < RK * TILE;      i += 32) WwT[i] = 0.f;
  __syncthreads();

  // ---- densify weights (atomicAdd folds mirror-duplicated taps) ----
  for (int idx = lane; idx < taps * TILE; idx += 32) {
    const int t = idx >> 4, j = idx & 15;
    const int r  = fov2[t * OUT_HW + oh0 + j];
    const int kk = min(max(r - r0, 0), RK - 1);
    atomicAdd(&Hw[j * HSTRIDE + kk], w2[t * OUT_HW + oh0 + j]);
    const int c  = fov3[t * OUT_HW + ow0 + j];
    const int cc = min(max(c - c0, 0), RK - 1);
    atomicAdd(&WwT[cc * TILE + j], w3[t * OUT_HW + ow0 + j]);
  }

#ifdef USE_ASYNC_LDS
  __builtin_amdgcn_s_wait_asynccnt(0);   // patch DMA landed in LDS
#else
  // ---- synchronous fallback: load 48x48 patch (zero past image edge) ----
  for (int idx = lane; idx < RK * RK; idx += 32) {
    const int rr = idx / RK, cc = idx - rr * RK;
    const int r = r0 + rr, c = c0 + cc;
    float v = 0.f;
    if (r < IN_HW && c < IN_HW) v = inp[r * IN_HW + c];
    P[idx] = v;
  }
#endif
  __syncthreads();

  // ---- pass 1: T(16x48) = Hw(16x48) @ P(48x48), three 16-wide N tiles ----
  v8f acc0 = {}, acc1 = {}, acc2 = {};
  #pragma unroll
  for (int k4 = 0; k4 < RK / 4; ++k4) {
    const int kb = k4 * 4 + 2 * g;       // this lane group's K pair
    v2f a;  a.x = Hw[lm * HSTRIDE + kb];  a.y = Hw[lm * HSTRIDE + kb + 1];
    v2f b0; b0.x = P[kb * RK +  0 + lm];  b0.y = P[(kb + 1) * RK +  0 + lm];
    v2f b1; b1.x = P[kb * RK + 16 + lm];  b1.y = P[(kb + 1) * RK + 16 + lm];
    v2f b2; b2.x = P[kb * RK + 32 + lm];  b2.y = P[(kb + 1) * RK + 32 + lm];
    acc0 = wmma_f32(a, b0, acc0);
    acc1 = wmma_f32(a, b1, acc1);
    acc2 = wmma_f32(a, b2, acc2);
  }
  // D layout: VGPR i -> row (i + 8*g), col lm. Spill T to LDS in A layout.
  #pragma unroll
  for (int i = 0; i < 8; ++i) {
    const int m = i + 8 * g;
    T[m * HSTRIDE +  0 + lm] = acc0[i];
    T[m * HSTRIDE + 16 + lm] = acc1[i];
    T[m * HSTRIDE + 32 + lm] = acc2[i];
  }
  __syncthreads();

  // ---- pass 2: OUT(16x16) = T(16x48) @ WwT(48x16) ----
  v8f acc = {};
  #pragma unroll
  for (int k4 = 0; k4 < RK / 4; ++k4) {
    const int kb = k4 * 4 + 2 * g;
    v2f a; a.x = T[lm * HSTRIDE + kb];   a.y = T[lm * HSTRIDE + kb + 1];
    v2f b; b.x = WwT[kb * TILE + lm];    b.y = WwT[(kb + 1) * TILE + lm];
    acc = wmma_f32(a, b, acc);
  }

  // ---- store: coalesced 16-lane rows per accumulator register ----
  float* op = out + ((size_t)p * OUT_HW + oh0) * OUT_HW + ow0;
  #pragma unroll
  for (int i = 0; i < 8; ++i) {
    op[(size_t)(i + 8 * g) * OUT_HW + lm] = acc[i];
  }
}

extern "C" void kernel_launch(void* const* d_in, const int* in_sizes, int n_in,
                              void* d_out, int out_size, void* d_ws, size_t ws_size,
                              hipStream_t stream) {
  const float* in   = (const float*)d_in[0];
  const float* w2   = (const float*)d_in[1];
  const int*   fov2 = (const int*)  d_in[2];
  const float* w3   = (const float*)d_in[3];
  const int*   fov3 = (const int*)  d_in[4];
  float* out = (float*)d_out;

  const int taps = in_sizes[1] / OUT_HW;   // w2 is (taps, 256, 1,1,1)

  const int tiles = PLANES * (OUT_HW / TILE) * (OUT_HW / TILE);  // 96*16*16
  dim3 grid(tiles), block(32);
  hipLaunchKernelGGL(resize_bicubic_fused, grid, block, 0, stream,
                     in, w2, fov2, w3, fov3, out, taps);
}